// HIGnnInterface_38156489458002
// MI455X (gfx1250) — compile-verified
//
#include <hip/hip_runtime.h>
#include <math.h>

typedef __attribute__((ext_vector_type(2))) float v2f;
typedef __attribute__((ext_vector_type(8))) float v8f;

#define HID   256
#define DE    64
#define CIN   192
#define NIMG  32768
#define NCLS  8192
#define NEDGE 131072
#define KCAT  (HID + DE)          // 320
#define INV_SQRT2 0.70710678118654752f
#define ALPHA_CAT 0.790569415042095f   // sqrt(640)/16 * 0.5
#define BETA_CAT  1.581138830084190f   // sqrt(640)/8  * 0.5
#define INV_SILU_C (1.0f / 0.596f)

__device__ __forceinline__ float wave_reduce_sum(float v) {
#pragma unroll
    for (int off = 16; off > 0; off >>= 1) v += __shfl_xor(v, off, 32);
    return v;
}

// ---------------------------------------------------------------------------
// Per-row geolin scale: sc[r] = (1/sqrt(fan)) / (EPS + ||w_r|| / sqrt(fan))
// One wave per row, 8 rows per block, HID=256 rows total.
// ---------------------------------------------------------------------------
__global__ __launch_bounds__(256) void geolin_scale_kernel(
    const float* __restrict__ W, int K, float* __restrict__ sc) {
  int row  = blockIdx.x * 8 + (threadIdx.x >> 5);
  int lane = threadIdx.x & 31;
  float s = 0.0f;
  const float* wp = W + (size_t)row * K;
  for (int k = lane; k < K; k += 32) { float w = wp[k]; s += w * w; }
  s = wave_reduce_sum(s);
  if (lane == 0) {
    float inv_sf = rsqrtf((float)K);
    sc[row] = inv_sf / (1e-4f + sqrtf(s) * inv_sf);
  }
}

// ---------------------------------------------------------------------------
// WMMA fp32 GEMM: Y[m, n] = outscale * colscale[n] * sum_k A(m,k) * W[n,k]
//   AMODE 0: A row-major [M,K]
//   AMODE 1: A = NCHW image tensor gather: A(m,k) = x[((m>>12)*CIN + k)*4096 + (m&4095)]
//   AMODE 2: A row-major [M,K] scaled by rcnt[m]  (scatter-mean)
//   CSTORE 0: Y row-major [M,HID]; CSTORE 1: NCHW scatter Y[((m>>12)*HID+n)*4096 + (m&4095)]
//   CADD: accumulate into existing Y (mp_sum fusion)
// One wave computes one 16x16 tile via v_wmma_f32_16x16x4_f32; 8 waves/block
// cover 128 output columns. grid = (M/16, HID/128).
// ---------------------------------------------------------------------------
template <int AMODE, int CSTORE, bool CADD>
__global__ __launch_bounds__(256) void wmma_gemm_kernel(
    const float* __restrict__ A, const float* __restrict__ W,
    const float* __restrict__ colscale, const float* __restrict__ rcnt,
    float* __restrict__ Y, int K, float outscale) {
  const int lane = threadIdx.x & 31;
  const int wave = threadIdx.x >> 5;
  const int half = lane >> 4;     // which K pair this lane holds
  const int lrow = lane & 15;
  const int m0 = blockIdx.x * 16;
  const int n0 = (blockIdx.y * 8 + wave) * 16;

  const int am = m0 + lrow;       // A row owned by this lane
  const int bn = n0 + lrow;       // weight row (output column)
  const float* wp = W + (size_t)bn * K + 2 * half;
  const float* ap = A + (size_t)am * K + 2 * half;

  float rc = 1.0f;
  if (AMODE == 2) rc = rcnt[am];
  int ab = 0, ahw = 0;
  if (AMODE == 1) { ab = am >> 12; ahw = am & 4095; }

  v8f acc = {};
#pragma unroll 8
  for (int k = 0; k < K; k += 4) {
    v2f afrag;
    if (AMODE == 1) {
      size_t base = ((size_t)(ab * CIN + k + 2 * half)) * 4096 + ahw;
      afrag.x = A[base];
      afrag.y = A[base + 4096];
    } else {
      afrag = *(const v2f*)(ap + k);
      if (AMODE == 2) { afrag.x *= rc; afrag.y *= rc; }
    }
    v2f bfrag = *(const v2f*)(wp + k);
    acc = __builtin_amdgcn_wmma_f32_16x16x4_f32(
        false, afrag, false, bfrag, (short)0, acc, false, false);
  }

  const int n = n0 + lrow;                  // C/D: col = lane & 15
  const float sc = colscale[n] * outscale;
#pragma unroll
  for (int r = 0; r < 8; ++r) {
    int m = m0 + r + 8 * half;              // C/D: row = r (+8 for upper half)
    float v = acc[r] * sc;
    if (CSTORE == 0) {
      size_t idx = (size_t)m * HID + n;
      if (CADD) v += Y[idx];
      Y[idx] = v;
    } else {
      size_t idx = ((size_t)((m >> 12) * HID + n)) * 4096 + (m & 4095);
      Y[idx] = v;
    }
  }
}

// ---------------------------------------------------------------------------
// mp_silu: y = (x * sigmoid(x)) / 0.596
// ---------------------------------------------------------------------------
__global__ __launch_bounds__(256) void mp_silu_kernel(
    const float* __restrict__ x, float* __restrict__ y, int n) {
  int i = blockIdx.x * 256 + threadIdx.x;
  if (i < n) {
    float v = x[i];
    y[i] = (v / (1.0f + __expf(-v))) * INV_SILU_C;
  }
}

// ---------------------------------------------------------------------------
// Edge scatter: agg[dst] += mp_cat(alpha * a_src[src], beta * ea[e]); cnt[dst]++
// One 320-thread block per edge (cols 0..255 = src feat, 256..319 = edge attr).
// ---------------------------------------------------------------------------
__global__ __launch_bounds__(320) void scatter_edges_kernel(
    const float* __restrict__ asrc, const float* __restrict__ ea,
    const int* __restrict__ src, const int* __restrict__ dst,
    float* __restrict__ agg, float* __restrict__ cnt) {
  int e = blockIdx.x;
  int t = threadIdx.x;
  int s = src[e], d = dst[e];
  float v;
  if (t < HID) v = ALPHA_CAT * asrc[(size_t)s * HID + t];
  else         v = BETA_CAT * ea[(size_t)e * DE + (t - HID)];
  atomicAdd(&agg[(size_t)d * KCAT + t], v);
  if (t == 0) atomicAdd(&cnt[d], 1.0f);
}

__global__ __launch_bounds__(256) void rcnt_kernel(float* __restrict__ cnt, int n) {
  int i = blockIdx.x * 256 + threadIdx.x;
  if (i < n) cnt[i] = 1.0f / fmaxf(cnt[i], 1.0f);
}

// ---------------------------------------------------------------------------
// Row L2 normalize (HID=256 cols): one wave per row, 8 rows per block.
// ---------------------------------------------------------------------------
__global__ __launch_bounds__(256) void rownorm_kernel(float* __restrict__ h, int M) {
  int row  = blockIdx.x * 8 + (threadIdx.x >> 5);
  int lane = threadIdx.x & 31;
  if (row >= M) return;
  float* p = h + (size_t)row * HID;
  float v[8], s = 0.0f;
#pragma unroll
  for (int j = 0; j < 8; ++j) { v[j] = p[lane + 32 * j]; s += v[j] * v[j]; }
  s = wave_reduce_sum(s);
  float inv = 1.0f / fmaxf(sqrtf(s), 1e-12f);
#pragma unroll
  for (int j = 0; j < 8; ++j) p[lane + 32 * j] = v[j] * inv;
}

// ---------------------------------------------------------------------------
extern "C" void kernel_launch(void* const* d_in, const int* in_sizes, int n_in,
                              void* d_out, int out_size, void* d_ws, size_t ws_size,
                              hipStream_t stream) {
  const float* x        = (const float*)d_in[0];
  const float* class_x  = (const float*)d_in[1];
  const float* ea_c2i   = (const float*)d_in[2];
  const float* ea_i2c   = (const float*)d_in[3];
  const float* Wm[12] = {
      (const float*)d_in[4],  (const float*)d_in[5],   // w_in_img, w_in_cls
      (const float*)d_in[6],  (const float*)d_in[7],   // wl_c2i_1, wr_c2i_1
      (const float*)d_in[8],  (const float*)d_in[9],   // wl_i2c_1, wr_i2c_1
      (const float*)d_in[10], (const float*)d_in[11],  // wl_c2i_2, wr_c2i_2
      (const float*)d_in[12], (const float*)d_in[13],  // wl_i2c_2, wr_i2c_2
      (const float*)d_in[14], (const float*)d_in[15]}; // w_out_img, w_out_cls
  const int fans[12] = {CIN, 128, KCAT, HID, KCAT, HID, KCAT, HID, KCAT, HID, HID, HID};
  const int* src_c2i = (const int*)d_in[16];
  const int* dst_c2i = (const int*)d_in[17];
  const int* src_i2c = (const int*)d_in[18];
  const int* dst_i2c = (const int*)d_in[19];

  // workspace carve-up (fp32)
  char* ws = (char*)d_ws;
  size_t off = 0;
  float* h_img   = (float*)(ws + off); off += (size_t)NIMG * HID * 4;   // 33.5 MB
  float* h_cls   = (float*)(ws + off); off += (size_t)NCLS * HID * 4;   //  8.4 MB
  float* a_img   = (float*)(ws + off); off += (size_t)NIMG * HID * 4;
  float* a_cls   = (float*)(ws + off); off += (size_t)NCLS * HID * 4;
  float* agg_img = (float*)(ws + off); off += (size_t)NIMG * KCAT * 4;  // 41.9 MB
  float* agg_cls = (float*)(ws + off); off += (size_t)NCLS * KCAT * 4;
  float* cnt_img = (float*)(ws + off); off += (size_t)NIMG * 4;
  float* cnt_cls = (float*)(ws + off); off += (size_t)NCLS * 4;
  float* scales  = (float*)(ws + off); off += (size_t)12 * HID * 4;

  float* out_img = (float*)d_out;                              // [8,256,64,64]
  float* out_cls = (float*)d_out + (size_t)8 * HID * 4096;     // [8192,256]

  // 1) geolin per-row weight scales
  for (int i = 0; i < 12; ++i)
    geolin_scale_kernel<<<HID / 8, 256, 0, stream>>>(Wm[i], fans[i], scales + i * HID);

  // 2) input projections
  wmma_gemm_kernel<1, 0, false><<<dim3(NIMG / 16, 2), 256, 0, stream>>>(
      x, Wm[0], scales + 0 * HID, nullptr, h_img, CIN, 1.0f);
  wmma_gemm_kernel<0, 0, false><<<dim3(NCLS / 16, 2), 256, 0, stream>>>(
      class_x, Wm[1], scales + 1 * HID, nullptr, h_cls, 128, 1.0f);

  // 3) two hetero-conv layers
  for (int li = 0; li < 2; ++li) {
    const int wb = 2 + li * 4;  // wl_c2i, wr_c2i, wl_i2c, wr_i2c

    mp_silu_kernel<<<(NIMG * HID + 255) / 256, 256, 0, stream>>>(h_img, a_img, NIMG * HID);
    mp_silu_kernel<<<(NCLS * HID + 255) / 256, 256, 0, stream>>>(h_cls, a_cls, NCLS * HID);

    hipMemsetAsync(agg_img, 0, (size_t)NIMG * KCAT * 4, stream);
    hipMemsetAsync(agg_cls, 0, (size_t)NCLS * KCAT * 4, stream);
    hipMemsetAsync(cnt_img, 0, (size_t)NIMG * 4, stream);
    hipMemsetAsync(cnt_cls, 0, (size_t)NCLS * 4, stream);

    scatter_edges_kernel<<<NEDGE, 320, 0, stream>>>(a_cls, ea_c2i, src_c2i, dst_c2i,
                                                    agg_img, cnt_img);
    scatter_edges_kernel<<<NEDGE, 320, 0, stream>>>(a_img, ea_i2c, src_i2c, dst_i2c,
                                                    agg_cls, cnt_cls);

    rcnt_kernel<<<(NIMG + 255) / 256, 256, 0, stream>>>(cnt_img, NIMG);
    rcnt_kernel<<<(NCLS + 255) / 256, 256, 0, stream>>>(cnt_cls, NCLS);

    // image nodes: (lin_l(mean-agg) + lin_r(a_img)) / sqrt(2), then L2 norm
    wmma_gemm_kernel<2, 0, false><<<dim3(NIMG / 16, 2), 256, 0, stream>>>(
        agg_img, Wm[wb + 0], scales + (wb + 0) * HID, cnt_img, h_img, KCAT, INV_SQRT2);
    wmma_gemm_kernel<0, 0, true><<<dim3(NIMG / 16, 2), 256, 0, stream>>>(
        a_img, Wm[wb + 1], scales + (wb + 1) * HID, nullptr, h_img, HID, INV_SQRT2);
    rownorm_kernel<<<NIMG / 8, 256, 0, stream>>>(h_img, NIMG);

    // class nodes
    wmma_gemm_kernel<2, 0, false><<<dim3(NCLS / 16, 2), 256, 0, stream>>>(
        agg_cls, Wm[wb + 2], scales + (wb + 2) * HID, cnt_cls, h_cls, KCAT, INV_SQRT2);
    wmma_gemm_kernel<0, 0, true><<<dim3(NCLS / 16, 2), 256, 0, stream>>>(
        a_cls, Wm[wb + 3], scales + (wb + 3) * HID, nullptr, h_cls, HID, INV_SQRT2);
    rownorm_kernel<<<NCLS / 8, 256, 0, stream>>>(h_cls, NCLS);
  }

  // 4) output projections (image path stores straight into NCHW layout)
  wmma_gemm_kernel<0, 1, false><<<dim3(NIMG / 16, 2), 256, 0, stream>>>(
      h_img, Wm[10], scales + 10 * HID, nullptr, out_img, HID, 1.0f);
  wmma_gemm_kernel<0, 0, false><<<dim3(NCLS / 16, 2), 256, 0, stream>>>(
      h_cls, Wm[11], scales + 11 * HID, nullptr, out_cls, HID, 1.0f);
  (void)in_sizes; (void)n_in; (void)out_size; (void)ws_size; (void)off;
}